// Model_83751862272728
// MI455X (gfx1250) — compile-verified
//
#include <hip/hip_runtime.h>
#include <hip/hip_bf16.h>
#include <math.h>

// Problem constants (from reference): B=256, S=512, T=131
#define B_   256
#define S_   512
#define T_   131
#define TP   144          // T padded to 9*16 (WMMA tile granularity)
#define KT   36           // TP / 4  (K-slices per step, WMMA K=4)
#define NT   9            // TP / 16 (N tiles)
#define ESTR 145          // LDS row stride for E (conflict-free B-frag reads)
#define QSTR 148          // LDS row stride for p/q
#define FSTR 132          // LDS row stride for staged feats (conflict-free D writes)
#define FCOLS 128         // columns staged via async b128 (cols 128..130 read direct)

typedef __attribute__((ext_vector_type(2))) float v2f;
typedef __attribute__((ext_vector_type(8))) float v8f;

// 32-bit LDS byte offset of a generic pointer into __shared__ (addrspacecast).
__device__ __forceinline__ unsigned lds_off(void* p) {
    return (unsigned)(unsigned long long)(__attribute__((address_space(3))) void*)p;
}

// ---------------------------------------------------------------------------
// Kernel A: E[i][j] = exp(cdt[types0[i][j], types1[i][j]]), zero in padding.
// Zero padding in exp-space == no contribution to the logsumexp.
// ---------------------------------------------------------------------------
__global__ void build_exp_trans(const int* __restrict__ t0,
                                const int* __restrict__ t1,
                                const float* __restrict__ cdt,
                                float* __restrict__ E) {
    int idx = blockIdx.x * blockDim.x + threadIdx.x;
    if (idx >= TP * TP) return;
    int i = idx / TP, j = idx % TP;
    float v = 0.0f;
    if (i < T_ && j < T_) {
        int a = t0[i * T_ + j];
        int b = t1[i * T_ + j];
        v = expf(cdt[a * 5 + b]);
    }
    E[idx] = v;
}

// ---------------------------------------------------------------------------
// Kernel B: gold path score per sequence (tiny work, 1 thread per batch row).
// ---------------------------------------------------------------------------
__global__ void gold_kernel(const float* __restrict__ feats,
                            const int* __restrict__ mask,
                            const int* __restrict__ tags,
                            const int* __restrict__ t0,
                            const int* __restrict__ t1,
                            const float* __restrict__ cdt,
                            const float* __restrict__ start,
                            const float* __restrict__ stop,
                            float* __restrict__ gold) {
    int b = blockIdx.x * blockDim.x + threadIdx.x;
    if (b >= B_) return;
    const size_t fb = (size_t)b * S_ * T_;
    int tg0 = tags[b * S_ + 0];
    float fs = feats[fb + tg0] * (float)mask[b * S_ + 0];
    float ts = 0.0f;
    int len = mask[b * S_ + 0];
    int prev = tg0;
    for (int s = 1; s < S_; ++s) {
        int tg = tags[b * S_ + s];
        float mf = (float)mask[b * S_ + s];
        fs += feats[fb + (size_t)s * T_ + tg] * mf;
        int ti = prev * T_ + tg;
        ts += cdt[t0[ti] * 5 + t1[ti]] * mf;
        len += mask[b * S_ + s];
        prev = tg;
    }
    int endtag = tags[b * S_ + (len - 1)];
    gold[b] = fs + ts + start[tg0] + stop[endtag];
}

// ---------------------------------------------------------------------------
// Kernel C: CRF forward recursion.  One workgroup = 16 batch rows for all S
// steps.  Per step:  m=rowmax(p); q=exp(p-m);  D = q (16xTP) x E (TPxTP) via
// v_wmma_f32_16x16x4_f32;  p[j] = mask ? m + log(D[j]) + feat[j] : p[j].
// 3 waves, each owns 3 of the 9 N-tiles.  Next step's feats are prefetched
// with double-buffered global_load_async_to_lds_b128 (ASYNCcnt path).
// ---------------------------------------------------------------------------
__launch_bounds__(96)
__global__ void crf_forward(const float* __restrict__ feats,
                            const int* __restrict__ mask,
                            const float* __restrict__ Eglob,
                            const float* __restrict__ start,
                            const float* __restrict__ stop,
                            const float* __restrict__ gold,
                            float* __restrict__ out) {
    __shared__ float E[TP][ESTR];       // exp(transitions), padded
    __shared__ float p[16][QSTR];       // log-domain partition
    __shared__ float q[16][QSTR];       // exp(p - rowmax)
    __shared__ float mrow[16];          // per-row max
    __shared__ float pmax[16][8];       // partial maxima (6 used)
    __shared__ float fbuf[2][16][FSTR]; // double-buffered feats stage (cols 0..127)

    const int tid  = threadIdx.x;
    const int lane = tid & 31;
    const int wave = tid >> 5;          // 0..2
    const int bm   = blockIdx.x * 16;   // batch tile base

    // Stage E into LDS (read many times per step).
    for (int idx = tid; idx < TP * TP; idx += 96)
        E[idx / TP][idx % TP] = Eglob[idx];

    // init: p[b][j] = feats[b,0,j] + start[j]
    for (int idx = tid; idx < 16 * TP; idx += 96) {
        int r = idx / TP, j = idx % TP;
        p[r][j] = (j < T_) ? feats[((size_t)(bm + r)) * S_ * T_ + j] + start[j]
                           : -INFINITY;
    }

    // kick off async staging of feats[., s=1, 0..127] into fbuf[1]
    {
        for (int g = tid; g < 512; g += 96) {          // 16 rows x 32 b128 chunks
            int row = g >> 5, ch = g & 31;
            const float* src = feats + ((size_t)(bm + row)) * S_ * T_
                             + (size_t)1 * T_ + ch * 4;
            unsigned dst = lds_off(&fbuf[1][row][ch * 4]);
            asm volatile("global_load_async_to_lds_b128 %0, %1, off"
                         :: "v"(dst), "v"(src) : "memory");
        }
    }
    __syncthreads();

    // WMMA fragment lane mapping (ISA 7.12.2, wave32):
    //  A 16x4 f32: lane holds row M=lane%16, K-pair kb..kb+1, kb = 2*(lane/16)
    //  B 4x16 f32: lane holds col N=lane%16, K rows kb..kb+1 (symmetric)
    //  C/D 16x16:  VGPR r -> row r (lanes 0-15) / row r+8 (lanes 16-31), col lane%16
    const int mA = lane & 15;
    const int kb = (lane >> 4) << 1;
    const int n0 = wave * 3;            // this wave's first N-tile

    for (int s = 1; s < S_; ++s) {
        const int cur = s & 1;

        // (0) issue async prefetch of feats for step s+1 into the other buffer
        if (s + 1 < S_) {
            const int nxt = (s + 1) & 1;
            for (int g = tid; g < 512; g += 96) {
                int row = g >> 5, ch = g & 31;
                const float* src = feats + ((size_t)(bm + row)) * S_ * T_
                                 + (size_t)(s + 1) * T_ + ch * 4;
                unsigned dst = lds_off(&fbuf[nxt][row][ch * 4]);
                asm volatile("global_load_async_to_lds_b128 %0, %1, off"
                             :: "v"(dst), "v"(src) : "memory");
            }
        }

        // (1) row max over valid states: 6 partial maxima per row, then reduce
        {
            int r = tid & 15, c = tid >> 4;            // c in 0..5
            int j0 = c * 22;
            int j1 = (c * 22 + 22 < T_) ? (c * 22 + 22) : T_;
            float m = -INFINITY;
            for (int j = j0; j < j1; ++j) m = fmaxf(m, p[r][j]);
            pmax[r][c] = m;
        }
        __syncthreads();
        if (tid < 16) {
            float m = pmax[tid][0];
            #pragma unroll
            for (int c = 1; c < 6; ++c) m = fmaxf(m, pmax[tid][c]);
            mrow[tid] = m;
        }
        __syncthreads();

        // (2) q = exp(p - m), zero in padding columns
        for (int idx = tid; idx < 16 * TP; idx += 96) {
            int r = idx / TP, j = idx % TP;
            q[r][j] = (j < T_) ? expf(p[r][j] - mrow[r]) : 0.0f;
        }
        __syncthreads();

        // (3) D = q x E via fp32 WMMA; A fragment reused across 3 N-tiles
        v8f acc0 = {}, acc1 = {}, acc2 = {};
        for (int k4 = 0; k4 < KT; ++k4) {
            const int kr = k4 * 4 + kb;
            v2f a;  a.x  = q[mA][kr];                 a.y  = q[mA][kr + 1];
            v2f b0; b0.x = E[kr][(n0    ) * 16 + mA]; b0.y = E[kr + 1][(n0    ) * 16 + mA];
            acc0 = __builtin_amdgcn_wmma_f32_16x16x4_f32(false, a, false, b0, (short)0, acc0, false, false);
            v2f b1; b1.x = E[kr][(n0 + 1) * 16 + mA]; b1.y = E[kr + 1][(n0 + 1) * 16 + mA];
            acc1 = __builtin_amdgcn_wmma_f32_16x16x4_f32(false, a, false, b1, (short)0, acc1, false, false);
            v2f b2; b2.x = E[kr][(n0 + 2) * 16 + mA]; b2.y = E[kr + 1][(n0 + 2) * 16 + mA];
            acc2 = __builtin_amdgcn_wmma_f32_16x16x4_f32(false, a, false, b2, (short)0, acc2, false, false);
        }

        // drain this step's staged feats (issued last step) and make visible
        asm volatile("s_wait_asynccnt 0x0" ::: "memory");
        __syncthreads();

        // (4) masked partition update: p = mask ? m + log(D) + feat : p
        #pragma unroll
        for (int t = 0; t < 3; ++t) {
            const int ntile = n0 + t;
            const v8f acc = (t == 0) ? acc0 : ((t == 1) ? acc1 : acc2);
            const int col = ntile * 16 + mA;
            if (col < T_) {
                #pragma unroll
                for (int r = 0; r < 8; ++r) {
                    const int row = (lane < 16) ? r : (r + 8);
                    if (mask[(bm + row) * S_ + s]) {
                        float d  = acc[r];
                        float ft = (col < FCOLS)
                                 ? fbuf[cur][row][col]
                                 : feats[((size_t)(bm + row)) * S_ * T_
                                         + (size_t)s * T_ + col];
                        p[row][col] = mrow[row] + logf(d) + ft;
                    }
                }
            }
        }
        __syncthreads();   // p visible for next rowmax; q/fbuf reads done before rewrite
    }

    // final: fwd = logsumexp_j(p[b][j] + stop[j]);  out = fwd - gold
    if (tid < 16) {
        float m = -INFINITY;
        for (int j = 0; j < T_; ++j) m = fmaxf(m, p[tid][j] + stop[j]);
        float ssum = 0.0f;
        for (int j = 0; j < T_; ++j) ssum += expf(p[tid][j] + stop[j] - m);
        out[bm + tid] = (m + logf(ssum)) - gold[bm + tid];
    }
}

// ---------------------------------------------------------------------------
extern "C" void kernel_launch(void* const* d_in, const int* in_sizes, int n_in,
                              void* d_out, int out_size, void* d_ws, size_t ws_size,
                              hipStream_t stream) {
    const float* feats = (const float*)d_in[0];
    const int*   mask  = (const int*)  d_in[1];
    const int*   tags  = (const int*)  d_in[2];   // jax default int64 -> i32 on device
    const int*   t0    = (const int*)  d_in[3];
    const int*   t1    = (const int*)  d_in[4];
    const float* cdt   = (const float*)d_in[5];
    const float* start = (const float*)d_in[6];
    const float* stop  = (const float*)d_in[7];
    float*       out   = (float*)d_out;

    float* E    = (float*)d_ws;          // TP*TP floats
    float* gold = E + TP * TP;           // B_ floats

    build_exp_trans<<<(TP * TP + 255) / 256, 256, 0, stream>>>(t0, t1, cdt, E);
    gold_kernel<<<(B_ + 127) / 128, 128, 0, stream>>>(feats, mask, tags, t0, t1,
                                                      cdt, start, stop, gold);
    crf_forward<<<B_ / 16, 96, 0, stream>>>(feats, mask, E, start, stop, gold, out);
}